// GraphGCN_32856499815217
// MI455X (gfx1250) — compile-verified
//
#include <hip/hip_runtime.h>

#define N_NODES   100000
#define N_EDGES   3200000
#define N_FEAT    10
#define HIDDEN    20
#define N_GRAPHS  512
#define N_CLASSES 2

typedef __attribute__((ext_vector_type(2))) float v2f;
typedef __attribute__((ext_vector_type(8))) float v8f;

// ---------------- utility ----------------
__global__ void fill_zero_k(float* __restrict__ p, int n) {
    int i = blockIdx.x * blockDim.x + threadIdx.x;
    if (i < n) p[i] = 0.0f;
}

// ---------------- degree / rsqrt ----------------
__global__ void deg_kernel(const int* __restrict__ col, const float* __restrict__ w,
                           float* __restrict__ deg, int E) {
    int e = blockIdx.x * blockDim.x + threadIdx.x;
    if (e < E) atomicAdd(&deg[col[e]], w[e]);
}

__global__ void dis_kernel(float* __restrict__ deg, int n) {
    int i = blockIdx.x * blockDim.x + threadIdx.x;
    if (i < n) deg[i] = rsqrtf(deg[i] + 1.0f);   // self-loop adds 1
}

// ---------------- dense matmul via V_WMMA_F32_16X16X4_F32 ----------------
// out[nrows,HIDDEN] = A[nrows,K] @ W[K,HIDDEN].  One WAVE computes a full
// 16-row x 20-col stripe (two 16x16 accumulators sharing the A fragment).
// 256-thread blocks = 8 waves = 8 row tiles per block; tile guard is
// wave-uniform so EXEC is all-ones at every WMMA.
// All loads are UNCONDITIONAL from clamped (in-bounds) addresses; the k/col
// zero-padding is applied as value masks (folds to v_cndmask, no branches).
// A frag: lanes 0-15 rows with K {0,1} in regs; lanes 16-31 same rows, K {2,3}.
// B frag: lanes = cols, same K striping.  C/D: reg r -> row r (lanes<16) / r+8.
template <int K>
__global__ __launch_bounds__(256) void matmul_wmma_f32(
        const float* __restrict__ A, const float* __restrict__ W,
        float* __restrict__ out, int ntiles) {
    constexpr int KITERS = (K + 3) / 4;

    const int lane = threadIdx.x & 31;
    const int wid  = threadIdx.x >> 5;
    const int tile = blockIdx.x * 8 + wid;
    if (tile >= ntiles) return;             // uniform per wave

    const int mlo   = lane & 15;
    const int khalf = (lane >> 4) * 2;      // 0 | 2
    const int arow  = tile * 16 + mlo;

    const int   ncol0  = mlo;               // cols 0..15 : always valid
    const int   ncol1  = 16 + mlo;          // cols 16..31: valid iff < HIDDEN
    const bool  c1ok   = ncol1 < HIDDEN;
    const int   ncol1c = c1ok ? ncol1 : (HIDDEN - 1);
    const float c1m    = c1ok ? 1.0f : 0.0f;

    const float* __restrict__ Arow = A + (size_t)arow * K;

    v8f acc0 = {};
    v8f acc1 = {};
#pragma unroll
    for (int kk = 0; kk < KITERS; ++kk) {
        const int k0 = kk * 4 + khalf;
        const int k1 = k0 + 1;
        const int k0c = (k0 < K) ? k0 : (K - 1);      // clamped, in-bounds
        const int k1c = (k1 < K) ? k1 : (K - 1);
        const float m0 = (k0 < K) ? 1.0f : 0.0f;      // value masks (no branches)
        const float m1 = (k1 < K) ? 1.0f : 0.0f;

        // unconditional loads from clamped addresses
        const float av0  = Arow[k0c];
        const float av1  = Arow[k1c];
        const float bv00 = W[k0c * HIDDEN + ncol0];
        const float bv01 = W[k1c * HIDDEN + ncol0];
        const float bv10 = W[k0c * HIDDEN + ncol1c];
        const float bv11 = W[k1c * HIDDEN + ncol1c];

        v2f a, b0, b1;
        a.x  = av0 * m0;
        a.y  = av1 * m1;
        b0.x = bv00 * m0;
        b0.y = bv01 * m1;
        b1.x = bv10 * (m0 * c1m);
        b1.y = bv11 * (m1 * c1m);

        // (neg_a, A, neg_b, B, c_mod, C, reuse_a, reuse_b)
        acc0 = __builtin_amdgcn_wmma_f32_16x16x4_f32(false, a, false, b0,
                                                     (short)0, acc0, false, false);
        acc1 = __builtin_amdgcn_wmma_f32_16x16x4_f32(false, a, false, b1,
                                                     (short)0, acc1, false, false);
    }

    const int rbase = tile * 16 + (lane >> 4) * 8;
#pragma unroll
    for (int r = 0; r < 8; ++r) {
        const size_t orow = (size_t)(rbase + r);
        out[orow * HIDDEN + ncol0] = acc0[r];
        if (c1ok) out[orow * HIDDEN + ncol1] = acc1[r];
    }
}

// ---------------- edge scatter:  agg[col] += xw[row] * norm ----------------
__global__ void scatter_kernel(const float* __restrict__ xw, const float* __restrict__ dis,
                               const int* __restrict__ row, const int* __restrict__ col,
                               const float* __restrict__ w, float* __restrict__ agg, int E) {
    int e = blockIdx.x * blockDim.x + threadIdx.x;
    if (e >= E) return;
    int r = row[e], c = col[e];
    float nrm = dis[r] * w[e] * dis[c];
    const float4* src = (const float4*)(xw + (size_t)r * HIDDEN);  // 80B rows -> 16B aligned
    float* dst = agg + (size_t)c * HIDDEN;
#pragma unroll
    for (int q = 0; q < 5; ++q) {
        float4 v = src[q];
        atomicAdd(&dst[q * 4 + 0], v.x * nrm);
        atomicAdd(&dst[q * 4 + 1], v.y * nrm);
        atomicAdd(&dst[q * 4 + 2], v.z * nrm);
        atomicAdd(&dst[q * 4 + 3], v.w * nrm);
    }
}

// ---------------- combine + L2 normalize + ReLU ----------------
__global__ void combine_kernel(const float* __restrict__ agg, const float* __restrict__ xw,
                               const float* __restrict__ dis, const float* __restrict__ bias,
                               float* __restrict__ hout, int n) {
    int v = blockIdx.x * blockDim.x + threadIdx.x;
    if (v >= n) return;
    float d = dis[v];
    float dd = d * d;
    float t[HIDDEN];
    float s = 0.0f;
#pragma unroll
    for (int h = 0; h < HIDDEN; ++h) {
        float val = agg[(size_t)v * HIDDEN + h] + xw[(size_t)v * HIDDEN + h] * dd + bias[h];
        t[h] = val;
        s += val * val;
    }
    float inv = 1.0f / fmaxf(sqrtf(s), 1e-12f);
#pragma unroll
    for (int h = 0; h < HIDDEN; ++h)
        hout[(size_t)v * HIDDEN + h] = fmaxf(t[h] * inv, 0.0f);
}

// ---------------- pooling (post-ReLU values >= 0, so int-bit max is valid) ----------------
__global__ void pool_kernel(const float* __restrict__ h, const int* __restrict__ batch,
                            float* __restrict__ sums, int* __restrict__ maxs,
                            int* __restrict__ cnt, int n) {
    int v = blockIdx.x * blockDim.x + threadIdx.x;
    if (v >= n) return;
    int g = batch[v];
    atomicAdd(&cnt[g], 1);
#pragma unroll
    for (int hh = 0; hh < HIDDEN; ++hh) {
        float val = h[(size_t)v * HIDDEN + hh];
        atomicAdd(&sums[g * HIDDEN + hh], val);
        atomicMax(&maxs[g * HIDDEN + hh], __float_as_int(val));
    }
}

// ---------------- final linear head ----------------
__global__ void final_kernel(const float* __restrict__ sums, const int* __restrict__ maxs,
                             const int* __restrict__ cnt, const float* __restrict__ linW,
                             const float* __restrict__ linb, float* __restrict__ out) {
    int idx = blockIdx.x * blockDim.x + threadIdx.x;
    if (idx >= N_GRAPHS * N_CLASSES) return;
    int g = idx / N_CLASSES, cls = idx % N_CLASSES;
    float invc = 1.0f / fmaxf((float)cnt[g], 1.0f);
    float acc = linb[cls];
#pragma unroll
    for (int hh = 0; hh < HIDDEN; ++hh) {
        float mx = __int_as_float(maxs[g * HIDDEN + hh]);            // maxv (0 if empty)
        float mn = sums[g * HIDDEN + hh] * invc;                     // meanv
        acc += mx * linW[hh * N_CLASSES + cls];
        acc += mn * linW[(HIDDEN + hh) * N_CLASSES + cls];
    }
    out[g * N_CLASSES + cls] = acc;
}

// ---------------- host orchestration ----------------
static inline int cdiv(int a, int b) { return (a + b - 1) / b; }

extern "C" void kernel_launch(void* const* d_in, const int* in_sizes, int n_in,
                              void* d_out, int out_size, void* d_ws, size_t ws_size,
                              hipStream_t stream) {
    (void)in_sizes; (void)n_in; (void)out_size; (void)ws_size;

    const float* x     = (const float*)d_in[0];
    const int*   eidx  = (const int*)  d_in[1];   // [2, E]
    const int*   batch = (const int*)  d_in[2];
    const float* ew    = (const float*)d_in[3];
    const float* W1    = (const float*)d_in[4];
    const float* b1    = (const float*)d_in[5];
    const float* W2    = (const float*)d_in[6];
    const float* b2    = (const float*)d_in[7];
    const float* W3    = (const float*)d_in[8];
    const float* b3    = (const float*)d_in[9];
    const float* linW  = (const float*)d_in[10];
    const float* linb  = (const float*)d_in[11];
    float* out = (float*)d_out;

    const int* row = eidx;
    const int* col = eidx + N_EDGES;

    // carve scratch (256B aligned)
    char* ws = (char*)d_ws;
    size_t off = 0;
    auto carve = [&](size_t bytes) -> char* {
        char* p = ws + off;
        off = (off + bytes + 255) & ~(size_t)255;
        return p;
    };
    float* dis  = (float*)carve((size_t)N_NODES * 4);             // deg -> dis in place
    float* xw   = (float*)carve((size_t)N_NODES * HIDDEN * 4);
    float* agg  = (float*)carve((size_t)N_NODES * HIDDEN * 4);
    float* hbuf = (float*)carve((size_t)N_NODES * HIDDEN * 4);
    float* sums = (float*)carve((size_t)N_GRAPHS * HIDDEN * 4);
    int*   maxs = (int*)  carve((size_t)N_GRAPHS * HIDDEN * 4);
    int*   cnt  = (int*)  carve((size_t)N_GRAPHS * 4);

    const int TB = 256;
    const int ROW_TILES = N_NODES / 16;        // 6250, exact
    const int MM_BLOCKS = cdiv(ROW_TILES, 8);  // 8 waves per block

    // degree + normalization factors
    fill_zero_k<<<cdiv(N_NODES, TB), TB, 0, stream>>>(dis, N_NODES);
    deg_kernel<<<cdiv(N_EDGES, TB), TB, 0, stream>>>(col, ew, dis, N_EDGES);
    dis_kernel<<<cdiv(N_NODES, TB), TB, 0, stream>>>(dis, N_NODES);

    // ---- layer 1 (K = N_FEAT) ----
    matmul_wmma_f32<N_FEAT><<<MM_BLOCKS, 256, 0, stream>>>(x, W1, xw, ROW_TILES);
    fill_zero_k<<<cdiv(N_NODES * HIDDEN, TB), TB, 0, stream>>>(agg, N_NODES * HIDDEN);
    scatter_kernel<<<cdiv(N_EDGES, TB), TB, 0, stream>>>(xw, dis, row, col, ew, agg, N_EDGES);
    combine_kernel<<<cdiv(N_NODES, TB), TB, 0, stream>>>(agg, xw, dis, b1, hbuf, N_NODES);

    // ---- layer 2 (K = HIDDEN) ----
    matmul_wmma_f32<HIDDEN><<<MM_BLOCKS, 256, 0, stream>>>(hbuf, W2, xw, ROW_TILES);
    fill_zero_k<<<cdiv(N_NODES * HIDDEN, TB), TB, 0, stream>>>(agg, N_NODES * HIDDEN);
    scatter_kernel<<<cdiv(N_EDGES, TB), TB, 0, stream>>>(xw, dis, row, col, ew, agg, N_EDGES);
    combine_kernel<<<cdiv(N_NODES, TB), TB, 0, stream>>>(agg, xw, dis, b2, hbuf, N_NODES);

    // ---- layer 3 (K = HIDDEN) ----
    matmul_wmma_f32<HIDDEN><<<MM_BLOCKS, 256, 0, stream>>>(hbuf, W3, xw, ROW_TILES);
    fill_zero_k<<<cdiv(N_NODES * HIDDEN, TB), TB, 0, stream>>>(agg, N_NODES * HIDDEN);
    scatter_kernel<<<cdiv(N_EDGES, TB), TB, 0, stream>>>(xw, dis, row, col, ew, agg, N_EDGES);
    combine_kernel<<<cdiv(N_NODES, TB), TB, 0, stream>>>(agg, xw, dis, b3, hbuf, N_NODES);

    // pooling
    fill_zero_k<<<cdiv(N_GRAPHS * HIDDEN, TB), TB, 0, stream>>>(sums, N_GRAPHS * HIDDEN);
    fill_zero_k<<<cdiv(N_GRAPHS * HIDDEN, TB), TB, 0, stream>>>((float*)maxs, N_GRAPHS * HIDDEN);
    fill_zero_k<<<cdiv(N_GRAPHS, TB), TB, 0, stream>>>((float*)cnt, N_GRAPHS);
    pool_kernel<<<cdiv(N_NODES, TB), TB, 0, stream>>>(hbuf, batch, sums, maxs, cnt, N_NODES);

    // head
    final_kernel<<<cdiv(N_GRAPHS * N_CLASSES, TB), TB, 0, stream>>>(sums, maxs, cnt, linW, linb, out);
}